// Morph2d_103079215602
// MI455X (gfx1250) — compile-verified
//
#include <hip/hip_runtime.h>
#include <stdint.h>
#include <math.h>

#define BLK    224                // 7 wave32s; 224 = 8 rows x 28 float4-chunks
#define HW     112
#define CHW    (HW * HW)          // 12544 floats per (b,c) plane
#define LDSW   120                // padded LDS row stride (floats); interior at col 4
#define LDSH   114                // 112 rows + 1 halo row top/bottom
#define NPASS  14                 // 112 rows / 8 rows-per-pass

typedef float v4f __attribute__((ext_vector_type(4)));   // builtin-legal 128-bit vector

__global__ __launch_bounds__(BLK) void morph2d_kernel(
    const float* __restrict__ x,     // [B,C,H,W]
    const float* __restrict__ kern,  // [1,C,3,3]
    const float* __restrict__ bias,  // [C]
    const float* __restrict__ pct,   // [C]
    float* __restrict__ out)         // [B,C,H,W]
{
    __shared__ float tile[LDSH * LDSW];   // 54,720 B

    const int bc  = blockIdx.x;           // 0 .. B*C-1
    const int c   = bc & 63;              // C = 64
    const int tid = threadIdx.x;

    const float* __restrict__ xp = x   + (size_t)bc * CHW;
    float*       __restrict__ op = out + (size_t)bc * CHW;

    // ---- zero the halo ring (disjoint from the async-load interior) ----
    // top (row 0) and bottom (row 113) padded rows, full stride (240 cells)
    for (int i = tid; i < 2 * LDSW; i += BLK) {
        int hi = (i >= LDSW);
        tile[(hi ? (LDSH - 1) * LDSW : 0) + (i - hi * LDSW)] = 0.0f;
    }
    // left (col 3) / right (col 116) halo columns, rows 1..112: one store/thread
    {
        int s  = (tid >= HW);
        int rr = 1 + (tid - s * HW);
        tile[rr * LDSW + (s ? (4 + HW) : 3)] = 0.0f;
    }

    // fixed per-thread mapping: chunk column q, starting row r0 (no in-loop div)
    const int r0 = tid / 28;              // 0..7
    const int q  = tid - r0 * 28;         // 0..27

    // ---- async DMA: interior 112x112 plane -> LDS (16B per lane per op) ----
    {
        const float* gp = xp + r0 * HW + q * 4;
        unsigned la = (unsigned)(uintptr_t)&tile[(r0 + 1) * LDSW + 4 + q * 4];
#pragma unroll
        for (int k = 0; k < NPASS; ++k) {
            asm volatile("global_load_async_to_lds_b128 %0, %1, off"
                         :
                         : "v"(la), "v"((unsigned long long)(uintptr_t)gp)
                         : "memory");
            gp += 8 * HW;                 // +3584 B global
            la += 8 * LDSW * 4;           // +3840 B LDS
        }
    }
    asm volatile("s_wait_asynccnt 0" ::: "memory");
    __syncthreads();

    // ---- per-channel uniforms (block-uniform -> scalar loads) ----
    const float bb = bias[c] * 10.0f;
    const float pv = 1.0f / (1.0f + __expf(-pct[c] * 10.0f));
    float kv[9], bk[9];
#pragma unroll
    for (int t = 0; t < 9; ++t) {
        kv[t] = kern[c * 9 + t];
        bk[t] = bb * kv[t];               // tap = fma(v, k, bb*k) == (v+bb)*k
    }

    // ---- stencil from LDS, float4 outputs, strength-reduced addressing ----
    {
        int   lbase = r0 * LDSW + q * 4 + 3;   // a[0][0] for padded row r0
        float* o    = op + r0 * HW + q * 4;

        for (int k = 0; k < NPASS; ++k) {
            float a[3][6];
#pragma unroll
            for (int i = 0; i < 3; ++i)
#pragma unroll
                for (int t = 0; t < 6; ++t)
                    a[i][t] = tile[lbase + i * LDSW + t];

            float r4[4];
#pragma unroll
            for (int jj = 0; jj < 4; ++jj) {
                float mx, mn;
#pragma unroll
                for (int i = 0; i < 3; ++i)
#pragma unroll
                    for (int j = 0; j < 3; ++j) {
                        float tap = fmaf(a[i][jj + j], kv[i * 3 + j], bk[i * 3 + j]);
                        if (i == 0 && j == 0) { mx = tap; mn = tap; }
                        else { mx = fmaxf(mx, tap); mn = fminf(mn, tap); }
                    }
                r4[jj] = fmaf(mx - mn, pv, mn) - bb;
            }

            v4f res = { r4[0], r4[1], r4[2], r4[3] };
            __builtin_nontemporal_store(res, (v4f*)o);     // streaming B128 store

            lbase += 8 * LDSW;
            o     += 8 * HW;
        }
    }
}

extern "C" void kernel_launch(void* const* d_in, const int* in_sizes, int n_in,
                              void* d_out, int out_size, void* d_ws, size_t ws_size,
                              hipStream_t stream) {
    const float* x    = (const float*)d_in[0];  // (16,64,112,112)
    const float* kern = (const float*)d_in[1];  // (1,64,3,3)
    const float* bias = (const float*)d_in[2];  // (64,)
    const float* pct  = (const float*)d_in[3];  // (64,)
    float* out = (float*)d_out;

    morph2d_kernel<<<dim3(16 * 64), dim3(BLK), 0, stream>>>(x, kern, bias, pct, out);
}